// Qwen3_5Layer_33028298506811
// MI455X (gfx1250) — compile-verified
//
#include <hip/hip_runtime.h>
#include <math.h>

typedef __attribute__((ext_vector_type(2))) float v2f;
typedef __attribute__((ext_vector_type(4))) float v4f;
typedef __attribute__((ext_vector_type(8))) float v8f;

#define B_   64
#define S_   2048
#define H_   2048
#define NH_  32
#define NKV_ 4
#define HD_  128
#define E_   32
#define I_   768
#define QKVN ((NH_ + 2 * NKV_) * HD_)   // 5120
#define GQ_  (NH_ / NKV_)               // 8

// ---------------------------------------------------------------------------
// RMSNorm: out[b,:] = x[b,:] * rsqrt(mean(x^2)+eps) * w
// ---------------------------------------------------------------------------
__global__ __launch_bounds__(256) void rmsnorm_kernel(const float* __restrict__ x,
                                                      const float* __restrict__ w,
                                                      float* __restrict__ out) {
  int b = blockIdx.x;
  const float* xb = x + (size_t)b * H_;
  float ss = 0.f;
  for (int i = threadIdx.x; i < H_; i += 256) { float v = xb[i]; ss += v * v; }
  for (int o = 16; o > 0; o >>= 1) ss += __shfl_xor(ss, o, 32);
  __shared__ float red[8];
  if ((threadIdx.x & 31) == 0) red[threadIdx.x >> 5] = ss;
  __syncthreads();
  float tot = 0.f;
#pragma unroll
  for (int i = 0; i < 8; ++i) tot += red[i];
  float sc = rsqrtf(tot * (1.0f / (float)H_) + 1e-6f);
  for (int i = threadIdx.x; i < H_; i += 256) out[(size_t)b * H_ + i] = xb[i] * sc * w[i];
}

// ---------------------------------------------------------------------------
// WMMA f32 GEMM, K-split: Cp[chunk][64,N] = A[64,Kc] @ B[N,Kc]^T  (partial)
// Wave computes a 64x32 stripe: 4 M-subtiles x 2 N-subtiles = 8 accumulators.
// Distance-2 software pipeline: two prefetch batches in flight, so WMMA waits
// are loadcnt<=6 (one batch), overlapping a full iteration of load latency.
// Weights (B) are single-use -> non-temporal loads keep L2 for activations.
// A 16x4 frag: lane -> (M = lane&15, K = k0 + 2*(lane>>4) + {0,1})  -> float2
// B 4x16 frag: lane -> (N = lane&15, K = k0 + 2*(lane>>4) + {0,1})  -> float2
// C/D 16x16:   vgpr v, lane -> (M = v + 8*(lane>>4), N = lane&15)
// ---------------------------------------------------------------------------
__global__ __launch_bounds__(256) void gemm_rowmajor_wmma(const float* __restrict__ A,
                                                          const float* __restrict__ Bm,
                                                          float* __restrict__ Cp,
                                                          int N, int K, int kchunks) {
  int wave = (int)((blockIdx.x * blockDim.x + threadIdx.x) >> 5);
  int lane = threadIdx.x & 31;
  int n0 = wave * 32;
  if (n0 >= N) return;
  int r = lane & 15, hi = lane >> 4;
  int chunk = blockIdx.y;
  int Kc = K / kchunks;
  int kbeg = chunk * Kc;
  int kend = kbeg + Kc;

  const float* bp0 = Bm + (size_t)(n0 + r) * K + 2 * hi;
  const float* bp1 = Bm + (size_t)(n0 + 16 + r) * K + 2 * hi;
  const float* ap0 = A + (size_t)(0  + r) * K + 2 * hi;
  const float* ap1 = A + (size_t)(16 + r) * K + 2 * hi;
  const float* ap2 = A + (size_t)(32 + r) * K + 2 * hi;
  const float* ap3 = A + (size_t)(48 + r) * K + 2 * hi;

  v8f c[8];
#pragma unroll
  for (int i = 0; i < 8; ++i) c[i] = (v8f){0.f,0.f,0.f,0.f,0.f,0.f,0.f,0.f};

  v2f ac[4], bc[2], an[4], bn[2], af[4], bf[2];
  bc[0] = __builtin_nontemporal_load((const v2f*)(bp0 + kbeg));
  bc[1] = __builtin_nontemporal_load((const v2f*)(bp1 + kbeg));
  ac[0] = *(const v2f*)(ap0 + kbeg);
  ac[1] = *(const v2f*)(ap1 + kbeg);
  ac[2] = *(const v2f*)(ap2 + kbeg);
  ac[3] = *(const v2f*)(ap3 + kbeg);
  bn[0] = __builtin_nontemporal_load((const v2f*)(bp0 + kbeg + 4));
  bn[1] = __builtin_nontemporal_load((const v2f*)(bp1 + kbeg + 4));
  an[0] = *(const v2f*)(ap0 + kbeg + 4);
  an[1] = *(const v2f*)(ap1 + kbeg + 4);
  an[2] = *(const v2f*)(ap2 + kbeg + 4);
  an[3] = *(const v2f*)(ap3 + kbeg + 4);

  for (int k0 = kbeg; k0 < kend - 8; k0 += 4) {
    int kf_ = k0 + 8;
    bf[0] = __builtin_nontemporal_load((const v2f*)(bp0 + kf_));
    bf[1] = __builtin_nontemporal_load((const v2f*)(bp1 + kf_));
    af[0] = *(const v2f*)(ap0 + kf_);
    af[1] = *(const v2f*)(ap1 + kf_);
    af[2] = *(const v2f*)(ap2 + kf_);
    af[3] = *(const v2f*)(ap3 + kf_);
#pragma unroll
    for (int mt = 0; mt < 4; ++mt)
#pragma unroll
      for (int nt = 0; nt < 2; ++nt)
        c[mt * 2 + nt] = __builtin_amdgcn_wmma_f32_16x16x4_f32(
            false, ac[mt], false, bc[nt], (short)0, c[mt * 2 + nt], false, false);
#pragma unroll
    for (int t = 0; t < 4; ++t) { ac[t] = an[t]; an[t] = af[t]; }
#pragma unroll
    for (int t = 0; t < 2; ++t) { bc[t] = bn[t]; bn[t] = bf[t]; }
  }
#pragma unroll
  for (int mt = 0; mt < 4; ++mt)
#pragma unroll
    for (int nt = 0; nt < 2; ++nt)
      c[mt * 2 + nt] = __builtin_amdgcn_wmma_f32_16x16x4_f32(
          false, ac[mt], false, bc[nt], (short)0, c[mt * 2 + nt], false, false);
#pragma unroll
  for (int mt = 0; mt < 4; ++mt)
#pragma unroll
    for (int nt = 0; nt < 2; ++nt)
      c[mt * 2 + nt] = __builtin_amdgcn_wmma_f32_16x16x4_f32(
          false, an[mt], false, bn[nt], (short)0, c[mt * 2 + nt], false, false);

  float* outp = Cp + (size_t)chunk * 64 * N;
#pragma unroll
  for (int mt = 0; mt < 4; ++mt)
#pragma unroll
    for (int nt = 0; nt < 2; ++nt)
#pragma unroll
      for (int v = 0; v < 8; ++v) {
        int row = mt * 16 + v + 8 * hi;
        int col = n0 + nt * 16 + r;
        outp[(size_t)row * N + col] = c[mt * 2 + nt][v];
      }
}

// ---------------------------------------------------------------------------
// Deterministic chunk reduction: out = sum_c part[c] (+ bias[n]) (+ resid)
// ---------------------------------------------------------------------------
__global__ __launch_bounds__(256) void reduce_chunks_kernel(const float* __restrict__ part,
                                                            int nchunks,
                                                            const float* __restrict__ bias,
                                                            const float* __restrict__ resid,
                                                            float* __restrict__ out, int N) {
  size_t idx = (size_t)blockIdx.x * 256 + threadIdx.x;  // over 64*N
  size_t MN = (size_t)64 * N;
  float s = 0.f;
  for (int cidx = 0; cidx < nchunks; ++cidx) s += part[(size_t)cidx * MN + idx];
  if (bias) s += bias[idx % N];
  if (resid) s += resid[idx];
  out[idx] = s;
}

// ---------------------------------------------------------------------------
// MoE down-proj GEMM, expert-split over grid.y:
// Cp[chunk][64,H] = sum_{e in chunk} act[:, e*I:(e+1)*I] @ w2[e]^T
// w2 is [E, H, I] row-major; same 64x32 stripe + distance-2 pipeline.
// ---------------------------------------------------------------------------
__global__ __launch_bounds__(256) void gemm_w2_wmma(const float* __restrict__ act,
                                                    const float* __restrict__ w2,
                                                    float* __restrict__ Cp,
                                                    int echunks) {
  const int N = H_;
  const int EI = E_ * I_;
  int wave = (int)((blockIdx.x * blockDim.x + threadIdx.x) >> 5);
  int lane = threadIdx.x & 31;
  int n0 = wave * 32;
  if (n0 >= N) return;
  int r = lane & 15, hi = lane >> 4;
  int epc = E_ / echunks;
  int ebeg = blockIdx.y * epc;
  int eend = ebeg + epc;

  v8f c[8];
#pragma unroll
  for (int i = 0; i < 8; ++i) c[i] = (v8f){0.f,0.f,0.f,0.f,0.f,0.f,0.f,0.f};

  for (int e = ebeg; e < eend; ++e) {
    const float* bp0 = w2 + ((size_t)e * N + (n0 + r)) * I_ + 2 * hi;
    const float* bp1 = w2 + ((size_t)e * N + (n0 + 16 + r)) * I_ + 2 * hi;
    const float* ap0 = act + (size_t)(0  + r) * EI + (size_t)e * I_ + 2 * hi;
    const float* ap1 = act + (size_t)(16 + r) * EI + (size_t)e * I_ + 2 * hi;
    const float* ap2 = act + (size_t)(32 + r) * EI + (size_t)e * I_ + 2 * hi;
    const float* ap3 = act + (size_t)(48 + r) * EI + (size_t)e * I_ + 2 * hi;

    v2f ac[4], bc[2], an[4], bn[2], af[4], bf[2];
    bc[0] = __builtin_nontemporal_load((const v2f*)(bp0));
    bc[1] = __builtin_nontemporal_load((const v2f*)(bp1));
    ac[0] = *(const v2f*)(ap0);
    ac[1] = *(const v2f*)(ap1);
    ac[2] = *(const v2f*)(ap2);
    ac[3] = *(const v2f*)(ap3);
    bn[0] = __builtin_nontemporal_load((const v2f*)(bp0 + 4));
    bn[1] = __builtin_nontemporal_load((const v2f*)(bp1 + 4));
    an[0] = *(const v2f*)(ap0 + 4);
    an[1] = *(const v2f*)(ap1 + 4);
    an[2] = *(const v2f*)(ap2 + 4);
    an[3] = *(const v2f*)(ap3 + 4);

    for (int i0 = 0; i0 < I_ - 8; i0 += 4) {
      int if_ = i0 + 8;
      bf[0] = __builtin_nontemporal_load((const v2f*)(bp0 + if_));
      bf[1] = __builtin_nontemporal_load((const v2f*)(bp1 + if_));
      af[0] = *(const v2f*)(ap0 + if_);
      af[1] = *(const v2f*)(ap1 + if_);
      af[2] = *(const v2f*)(ap2 + if_);
      af[3] = *(const v2f*)(ap3 + if_);
#pragma unroll
      for (int mt = 0; mt < 4; ++mt)
#pragma unroll
        for (int nt = 0; nt < 2; ++nt)
          c[mt * 2 + nt] = __builtin_amdgcn_wmma_f32_16x16x4_f32(
              false, ac[mt], false, bc[nt], (short)0, c[mt * 2 + nt], false, false);
#pragma unroll
      for (int t = 0; t < 4; ++t) { ac[t] = an[t]; an[t] = af[t]; }
#pragma unroll
      for (int t = 0; t < 2; ++t) { bc[t] = bn[t]; bn[t] = bf[t]; }
    }
#pragma unroll
    for (int mt = 0; mt < 4; ++mt)
#pragma unroll
      for (int nt = 0; nt < 2; ++nt)
        c[mt * 2 + nt] = __builtin_amdgcn_wmma_f32_16x16x4_f32(
            false, ac[mt], false, bc[nt], (short)0, c[mt * 2 + nt], false, false);
#pragma unroll
    for (int mt = 0; mt < 4; ++mt)
#pragma unroll
      for (int nt = 0; nt < 2; ++nt)
        c[mt * 2 + nt] = __builtin_amdgcn_wmma_f32_16x16x4_f32(
            false, an[mt], false, bn[nt], (short)0, c[mt * 2 + nt], false, false);
  }

  float* outp = Cp + (size_t)blockIdx.y * 64 * N;
#pragma unroll
  for (int mt = 0; mt < 4; ++mt)
#pragma unroll
    for (int nt = 0; nt < 2; ++nt)
#pragma unroll
      for (int v = 0; v < 8; ++v) {
        int row = mt * 16 + v + 8 * hi;
        int col = n0 + nt * 16 + r;
        outp[(size_t)row * N + col] = c[mt * 2 + nt][v];
      }
}

// ---------------------------------------------------------------------------
// GQA flash-decode per (b, kv-head). Block = 256 threads (8 waves); each wave
// handles all 8 query heads for a strided subset of s in [0, L]; s == L uses
// k_new/v_new from the qkv buffer (cache is NOT mutated). LDS combine.
// K/V cache rows are single-use -> non-temporal loads.
// ---------------------------------------------------------------------------
__global__ __launch_bounds__(256) void attn_decode_kernel(const float* __restrict__ qkv,
                                                          const float* __restrict__ kc,
                                                          const float* __restrict__ vc,
                                                          const int* __restrict__ seq_lens,
                                                          float* __restrict__ attn_out) {
  const float scale = 0.088388347648318447f;  // 1/sqrt(128)
  int b  = blockIdx.x / NKV_;
  int kv = blockIdx.x % NKV_;
  int wave = threadIdx.x >> 5;
  int lane = threadIdx.x & 31;
  int L = seq_lens[b];

  const float* qbase = qkv + (size_t)b * QKVN;
  v4f qf[GQ_];
#pragma unroll
  for (int g = 0; g < GQ_; ++g)
    qf[g] = *(const v4f*)(qbase + (kv * GQ_ + g) * HD_ + lane * 4);

  float m[GQ_], l[GQ_];
  v4f acc[GQ_];
#pragma unroll
  for (int g = 0; g < GQ_; ++g) { m[g] = -1e30f; l[g] = 0.f; acc[g] = (v4f){0.f, 0.f, 0.f, 0.f}; }

  const float* knew = qbase + NH_ * HD_ + kv * HD_;
  const float* vnew = qbase + (NH_ + NKV_) * HD_ + kv * HD_;

  for (int s = wave; s <= L; s += 8) {
    v4f kf, vf;
    if (s == L) {
      kf = *(const v4f*)(knew + lane * 4);
      vf = *(const v4f*)(vnew + lane * 4);
    } else {
      kf = __builtin_nontemporal_load((const v4f*)(kc + (((size_t)b * S_ + s) * NKV_ + kv) * HD_ + lane * 4));
      vf = __builtin_nontemporal_load((const v4f*)(vc + (((size_t)b * S_ + s) * NKV_ + kv) * HD_ + lane * 4));
    }
    float dot[GQ_];
#pragma unroll
    for (int g = 0; g < GQ_; ++g)
      dot[g] = qf[g].x * kf.x + qf[g].y * kf.y + qf[g].z * kf.z + qf[g].w * kf.w;
#pragma unroll
    for (int o = 16; o > 0; o >>= 1)
#pragma unroll
      for (int g = 0; g < GQ_; ++g) dot[g] += __shfl_xor(dot[g], o, 32);
#pragma unroll
    for (int g = 0; g < GQ_; ++g) {
      float sc = dot[g] * scale;
      float mn = fmaxf(m[g], sc);
      float corr = __expf(m[g] - mn);
      float p = __expf(sc - mn);
      l[g] = l[g] * corr + p;
      acc[g] = acc[g] * corr + p * vf;
      m[g] = mn;
    }
  }

  __shared__ float sm_m[8][GQ_];
  __shared__ float sm_l[8][GQ_];
  __shared__ float sm_acc[8][GQ_][HD_];
#pragma unroll
  for (int g = 0; g < GQ_; ++g) {
    if (lane == 0) { sm_m[wave][g] = m[g]; sm_l[wave][g] = l[g]; }
    *(v4f*)(&sm_acc[wave][g][lane * 4]) = acc[g];
  }
  __syncthreads();

  // wave w combines partials for head g = w
  int g = wave;
  float mstar = -1e30f;
#pragma unroll
  for (int i = 0; i < 8; ++i) mstar = fmaxf(mstar, sm_m[i][g]);
  float lsum = 0.f;
  v4f a = (v4f){0.f, 0.f, 0.f, 0.f};
#pragma unroll
  for (int i = 0; i < 8; ++i) {
    float f = __expf(sm_m[i][g] - mstar);
    lsum += sm_l[i][g] * f;
    v4f pa = *(const v4f*)(&sm_acc[i][g][lane * 4]);
    a = a + pa * f;
  }
  float inv = 1.0f / lsum;
  *(v4f*)(attn_out + (size_t)b * (NH_ * HD_) + (kv * GQ_ + g) * HD_ + lane * 4) = a * inv;
}

// ---------------------------------------------------------------------------
// Gating: logits = h2 @ gate_w^T; top-4 of logits, softmax over selected
// (== renormalized top-k of full softmax). route[b,e] dense [B,E].
// ---------------------------------------------------------------------------
__global__ __launch_bounds__(256) void gate_topk_kernel(const float* __restrict__ h2,
                                                        const float* __restrict__ gate_w,
                                                        float* __restrict__ route) {
  int b = blockIdx.x;
  int wave = threadIdx.x >> 5;
  int lane = threadIdx.x & 31;
  __shared__ float logits[E_];

  const float* hb = h2 + (size_t)b * H_;
  int e0 = wave * 4;
  float part[4] = {0.f, 0.f, 0.f, 0.f};
  for (int k = lane; k < H_; k += 32) {
    float hv = hb[k];
#pragma unroll
    for (int t = 0; t < 4; ++t) part[t] += hv * gate_w[(size_t)(e0 + t) * H_ + k];
  }
#pragma unroll
  for (int o = 16; o > 0; o >>= 1)
#pragma unroll
    for (int t = 0; t < 4; ++t) part[t] += __shfl_xor(part[t], o, 32);
  if (lane == 0)
#pragma unroll
    for (int t = 0; t < 4; ++t) logits[e0 + t] = part[t];
  __syncthreads();

  if (threadIdx.x == 0) {
    int sel[4];
    float selv[4];
    unsigned used = 0u;
#pragma unroll
    for (int t = 0; t < 4; ++t) {
      float best = -1e30f; int bi = 0;
      for (int e = 0; e < E_; ++e)
        if (!((used >> e) & 1u) && logits[e] > best) { best = logits[e]; bi = e; }
      sel[t] = bi; selv[t] = best; used |= (1u << bi);
    }
    float mx = selv[0];
    float w[4], s = 0.f;
#pragma unroll
    for (int t = 0; t < 4; ++t) { w[t] = __expf(selv[t] - mx); s += w[t]; }
    float invs = 1.0f / s;
    for (int e = 0; e < E_; ++e) route[(size_t)b * E_ + e] = 0.f;
#pragma unroll
    for (int t = 0; t < 4; ++t) route[(size_t)b * E_ + sel[t]] = w[t] * invs;
  }
}

// ---------------------------------------------------------------------------
// act[b,e,i] = route[b,e] * silu(gu[b,e,i]) * gu[b,e,I+i]
// ---------------------------------------------------------------------------
__global__ __launch_bounds__(256) void moe_act_kernel(const float* __restrict__ gu,
                                                      const float* __restrict__ route,
                                                      float* __restrict__ act) {
  size_t idx = (size_t)blockIdx.x * 256 + threadIdx.x;  // over B*E*I
  int i = (int)(idx % I_);
  size_t be = idx / I_;
  int e = (int)(be % E_);
  int b = (int)(be / E_);
  const float* g2 = gu + (size_t)b * (E_ * 2 * I_) + (size_t)e * (2 * I_);
  float g = g2[i];
  float u = g2[I_ + i];
  float rt = route[(size_t)b * E_ + e];
  float sig = 1.0f / (1.0f + __expf(-g));
  act[idx] = rt * (g * sig) * u;
}

// ---------------------------------------------------------------------------
extern "C" void kernel_launch(void* const* d_in, const int* in_sizes, int n_in,
                              void* d_out, int out_size, void* d_ws, size_t ws_size,
                              hipStream_t stream) {
  (void)in_sizes; (void)n_in; (void)out_size; (void)ws_size;
  const float* hidden  = (const float*)d_in[0];
  // d_in[1] = positions (unused, no RoPE in reference)
  const float* k_cache = (const float*)d_in[2];
  const float* v_cache = (const float*)d_in[3];
  const int*   seqlens = (const int*)d_in[4];
  const float* w_qkv   = (const float*)d_in[5];
  const float* b_qkv   = (const float*)d_in[6];
  const float* w_o     = (const float*)d_in[7];
  const float* ln1_w   = (const float*)d_in[8];
  const float* ln2_w   = (const float*)d_in[9];
  const float* gate_w  = (const float*)d_in[10];
  const float* w1      = (const float*)d_in[11];
  const float* w2      = (const float*)d_in[12];
  float* out = (float*)d_out;

  float* ws    = (float*)d_ws;
  float* h1    = ws;                                   // B*H
  float* qkv   = h1    + (size_t)B_ * H_;              // B*5120
  float* attn  = qkv   + (size_t)B_ * QKVN;            // B*4096
  float* hs2   = attn  + (size_t)B_ * NH_ * HD_;       // B*H
  float* h2    = hs2   + (size_t)B_ * H_;              // B*H
  float* route = h2    + (size_t)B_ * H_;              // B*E
  float* gu    = route + (size_t)B_ * E_;              // B*E*2I
  float* act   = gu    + (size_t)B_ * E_ * 2 * I_;     // B*E*I
  float* gpart = act   + (size_t)B_ * E_ * I_;         // up to 8*64*5120 floats

  // 1) pre-attn RMSNorm
  rmsnorm_kernel<<<B_, 256, 0, stream>>>(hidden, ln1_w, h1);
  // 2) fused QKV projection: 4 K-chunks (640 waves), then reduce + bias
  gemm_rowmajor_wmma<<<dim3(QKVN / 32 / 8, 4), 256, 0, stream>>>(h1, w_qkv, gpart, QKVN, H_, 4);
  reduce_chunks_kernel<<<(B_ * QKVN) / 256, 256, 0, stream>>>(gpart, 4, b_qkv, nullptr, qkv, QKVN);
  // 3) GQA paged-attention decode (cache slot L substituted, cache untouched)
  attn_decode_kernel<<<B_ * NKV_, 256, 0, stream>>>(qkv, k_cache, v_cache, seqlens, attn);
  // 4) o-proj: 8 K-chunks (512 waves), reduce + hidden residual -> hs2
  gemm_rowmajor_wmma<<<dim3(H_ / 32 / 8, 8), 256, 0, stream>>>(attn, w_o, gpart, H_, NH_ * HD_, 8);
  reduce_chunks_kernel<<<(B_ * H_) / 256, 256, 0, stream>>>(gpart, 8, nullptr, hidden, hs2, H_);
  // 5) post-attn RMSNorm
  rmsnorm_kernel<<<B_, 256, 0, stream>>>(hs2, ln2_w, h2);
  // 6) router: top-4 softmax weights
  gate_topk_kernel<<<B_, 256, 0, stream>>>(h2, gate_w, route);
  // 7) gate/up projections for all experts: gu = h2 @ w1^T (1536 waves, direct)
  gemm_rowmajor_wmma<<<dim3((E_ * 2 * I_) / 32 / 8, 1), 256, 0, stream>>>(h2, w1, gu, E_ * 2 * I_, H_, 1);
  // 8) route-scaled SwiGLU activation
  moe_act_kernel<<<(B_ * E_ * I_) / 256, 256, 0, stream>>>(gu, route, act);
  // 9) down projection: 4 expert-chunks, reduce + hs2 residual -> output
  gemm_w2_wmma<<<dim3(H_ / 32 / 8, 4), 256, 0, stream>>>(act, w2, gpart, 4);
  reduce_chunks_kernel<<<(B_ * H_) / 256, 256, 0, stream>>>(gpart, 4, nullptr, hs2, out, H_);
}